// MultiHeadAttentionLayer_40295383171716
// MI455X (gfx1250) — compile-verified
//
#include <hip/hip_runtime.h>

// ---------------------------------------------------------------------------
// Graph multi-head attention (H=8, D=16, IN_DIM=128) for MI455X / gfx1250.
// fp32 throughout; GEMMs use V_WMMA_F32_16X16X4_F32 (exact f32 matrix pipe).
// One wave = 16 rows x all 128 cols (A read once; 8 WMMA accumulators).
// Edge-score pass fused into the e-projection GEMM epilogue.
// Segment softmax via order-preserving-uint atomicMax + global_atomic_add_f32.
// ---------------------------------------------------------------------------

#define HD  128   // H*D == IN_DIM == 128
#define WPB 8     // waves per block

typedef float v2f __attribute__((ext_vector_type(2)));
typedef float v8f __attribute__((ext_vector_type(8)));

// Order-preserving map f32 -> u32 (monotonic): enables atomicMax on uint.
__device__ __forceinline__ unsigned f32_key(float x) {
    unsigned u = __float_as_uint(x);
    return (u & 0x80000000u) ? ~u : (u | 0x80000000u);
}
__device__ __forceinline__ float key_f32(unsigned k) {
    unsigned u = (k & 0x80000000u) ? (k & 0x7FFFFFFFu) : ~k;
    return __uint_as_float(u);
}

// f32 WMMA operand layout (ISA 7.12.2):
//   A 16x4:  lane%16 = M row; VGPR index = K&1; lane/16 = K>>1
//   B 4x16:  lane%16 = N col; VGPR index = K&1; lane/16 = K>>1
//   C 16x16: VGPR i -> M = i + 8*(lane/16), N = lane%16

// ---------------------------------------------------------------------------
// C[M,128] = A[M,128] * W[128,128] + bias[128]
// One wave: 16 rows x 128 cols. A fragment loaded once per k-step, used by
// 8 WMMAs into independent accumulators (no D->A/B hazard between them).
// ---------------------------------------------------------------------------
__global__ void gemm128_bias_wmma(const float* __restrict__ A,
                                  const float* __restrict__ W,
                                  const float* __restrict__ bias,
                                  float* __restrict__ C,
                                  int nRowTiles)
{
    const int lane    = threadIdx.x & 31;
    const int rowTile = blockIdx.x * WPB + (threadIdx.x >> 5);
    if (rowTile >= nRowTiles) return;          // wave-uniform; EXEC stays full

    const int l16   = lane & 15;
    const int khalf = (lane >> 4) << 1;        // 0 or 2 (K MSB from lane half)
    const float* arow = A + (size_t)(rowTile * 16 + l16) * HD;

    v8f acc[8] = {};
#pragma unroll 2
    for (int kk = 0; kk < HD; kk += 4) {
        v2f a;
        a.x = arow[kk + khalf];
        a.y = arow[kk + khalf + 1];
        const float* w0 = W + (size_t)(kk + khalf) * HD;
        const float* w1 = w0 + HD;
#pragma unroll
        for (int j = 0; j < 8; ++j) {
            v2f b;
            b.x = w0[j * 16 + l16];
            b.y = w1[j * 16 + l16];
            acc[j] = __builtin_amdgcn_wmma_f32_16x16x4_f32(
                         false, a, false, b, (short)0, acc[j], false, false);
        }
    }

    const int rbase = rowTile * 16 + ((lane >> 4) << 3);
#pragma unroll
    for (int j = 0; j < 8; ++j) {
        const int   col = j * 16 + l16;
        const float bv  = bias[col];
#pragma unroll
        for (int i = 0; i < 8; ++i)
            C[(size_t)(rbase + i) * HD + col] = acc[j][i] + bv;
    }
}

// ---------------------------------------------------------------------------
// Fused: e_out[16 edges,128] = e[16,128] @ We + be  + (K[src,h,:].Q[dst,h,:])/4
// plus atomicMax of the score into mkey[dst].  Col tile j == head j.
// Per-edge head dots computed by gathering K/Q rows (L2-resident tables),
// reduced with wave32 shuffles, staged through 512 B of LDS per wave (4KB per block).
// ---------------------------------------------------------------------------
__global__ void gemm_edge_score_fused(const float* __restrict__ Ae,   // e [E,128]
                                      const float* __restrict__ W,
                                      const float* __restrict__ bias,
                                      const float* __restrict__ Q,
                                      const float* __restrict__ K,
                                      const int*   __restrict__ src,
                                      const int*   __restrict__ dst,
                                      float*       __restrict__ e_out,
                                      unsigned*    __restrict__ mkey,
                                      int nRowTiles)
{
    __shared__ float sdots[WPB][16][8];        // [wave][edge-in-tile][head]

    const int lane    = threadIdx.x & 31;
    const int wv      = threadIdx.x >> 5;
    const int rowTile = blockIdx.x * WPB + wv;
    if (rowTile >= nRowTiles) return;          // E multiple of 128 -> no exits

    const int l16   = lane & 15;
    const int khalf = (lane >> 4) << 1;
    const float* arow = Ae + (size_t)(rowTile * 16 + l16) * HD;

    // --- projection GEMM ---
    v8f acc[8] = {};
#pragma unroll 2
    for (int kk = 0; kk < HD; kk += 4) {
        v2f a;
        a.x = arow[kk + khalf];
        a.y = arow[kk + khalf + 1];
        const float* w0 = W + (size_t)(kk + khalf) * HD;
        const float* w1 = w0 + HD;
#pragma unroll
        for (int j = 0; j < 8; ++j) {
            v2f b;
            b.x = w0[j * 16 + l16];
            b.y = w1[j * 16 + l16];
            acc[j] = __builtin_amdgcn_wmma_f32_16x16x4_f32(
                         false, a, false, b, (short)0, acc[j], false, false);
        }
    }

    // --- per-edge, per-head dots:  K[src].Q[dst] over D=16 ---
    const int myS = src[rowTile * 16 + l16];   // lanes 16-31 mirror 0-15
    const int myD = dst[rowTile * 16 + l16];
#pragma unroll 4
    for (int r = 0; r < 16; ++r) {
        const int s_r = __shfl(myS, r, 32);
        const int d_r = __shfl(myD, r, 32);
        const float4 kv = *(const float4*)(K + (size_t)s_r * HD + lane * 4);
        const float4 qv = *(const float4*)(Q + (size_t)d_r * HD + lane * 4);
        float p = kv.x * qv.x + kv.y * qv.y + kv.z * qv.z + kv.w * qv.w;
        p += __shfl_xor(p, 1, 32);             // reduce 4 lanes of one head
        p += __shfl_xor(p, 2, 32);
        if ((lane & 3) == 0)
            sdots[wv][r][lane >> 2] = p * 0.25f;   // 1/sqrt(D), D=16
    }
    __syncthreads();

    // --- epilogue: score = proj + bias + dot, store + atomic max ---
    const int half8 = (lane >> 4) << 3;        // 0 or 8
#pragma unroll
    for (int j = 0; j < 8; ++j) {
        const int   col = j * 16 + l16;
        const float bv  = bias[col];
#pragma unroll
        for (int i = 0; i < 8; ++i) {
            const int   r  = i + half8;        // edge index within tile
            const float sc = acc[j][i] + bv + sdots[wv][r][j];
            e_out[(size_t)(rowTile * 16 + r) * HD + col] = sc;
            const int d_r = __shfl(myD, r, 32);
            atomicMax(mkey + (size_t)d_r * HD + col, f32_key(sc));
        }
    }
}

// ---------------------------------------------------------------------------
// Pass 2: denom[dst] += exp(score - m[dst]).  One wave per edge.
// ---------------------------------------------------------------------------
__global__ void edge_denom_kernel(const float*    __restrict__ e_out,
                                  const unsigned* __restrict__ mkey,
                                  const int*      __restrict__ dst,
                                  float*          __restrict__ denom,
                                  int E)
{
    const int wid = blockIdx.x * (blockDim.x >> 5) + (threadIdx.x >> 5);
    if (wid >= E) return;
    const int lane = threadIdx.x & 31;
    const int d = dst[wid];

    const float4    sc = *(const float4*)(e_out + (size_t)wid * HD + lane * 4);
    const unsigned* mk = mkey  + (size_t)d * HD + lane * 4;
    float*          dn = denom + (size_t)d * HD + lane * 4;
    atomicAdd(dn + 0, __expf(sc.x - key_f32(mk[0])));
    atomicAdd(dn + 1, __expf(sc.y - key_f32(mk[1])));
    atomicAdd(dn + 2, __expf(sc.z - key_f32(mk[2])));
    atomicAdd(dn + 3, __expf(sc.w - key_f32(mk[3])));
}

// ---------------------------------------------------------------------------
// Pass 3: wV[dst] += V[src] * exp(score - m[dst]) / denom[dst]
// ---------------------------------------------------------------------------
__global__ void edge_wv_kernel(const float*    __restrict__ e_out,
                               const unsigned* __restrict__ mkey,
                               const float*    __restrict__ denom,
                               const float*    __restrict__ V,
                               const int*      __restrict__ src,
                               const int*      __restrict__ dst,
                               float*          __restrict__ wV,
                               int E)
{
    const int wid = blockIdx.x * (blockDim.x >> 5) + (threadIdx.x >> 5);
    if (wid >= E) return;
    const int lane = threadIdx.x & 31;
    const int s = src[wid];
    const int d = dst[wid];

    const float4    sc = *(const float4*)(e_out + (size_t)wid * HD + lane * 4);
    const float4    vv = *(const float4*)(V     + (size_t)s   * HD + lane * 4);
    const unsigned* mk = mkey  + (size_t)d * HD + lane * 4;
    const float*    dn = denom + (size_t)d * HD + lane * 4;
    float*          wp = wV    + (size_t)d * HD + lane * 4;
    atomicAdd(wp + 0, vv.x * __expf(sc.x - key_f32(mk[0])) / dn[0]);
    atomicAdd(wp + 1, vv.y * __expf(sc.y - key_f32(mk[1])) / dn[1]);
    atomicAdd(wp + 2, vv.z * __expf(sc.z - key_f32(mk[2])) / dn[2]);
    atomicAdd(wp + 3, vv.w * __expf(sc.w - key_f32(mk[3])) / dn[3]);
}

// ---------------------------------------------------------------------------
extern "C" void kernel_launch(void* const* d_in, const int* in_sizes, int n_in,
                              void* d_out, int out_size, void* d_ws, size_t ws_size,
                              hipStream_t stream)
{
    const float* h  = (const float*)d_in[0];
    const float* e  = (const float*)d_in[1];
    const float* Wq = (const float*)d_in[2];
    const float* bq = (const float*)d_in[3];
    const float* Wk = (const float*)d_in[4];
    const float* bk = (const float*)d_in[5];
    const float* Wv = (const float*)d_in[6];
    const float* bv = (const float*)d_in[7];
    const float* We = (const float*)d_in[8];
    const float* be = (const float*)d_in[9];
    const int*   src = (const int*)d_in[10];
    const int*   dst = (const int*)d_in[11];

    const int N = in_sizes[0] / HD;   // 50000
    const int E = in_sizes[1] / HD;   // 800000

    float* out   = (float*)d_out;
    float* wV    = out;                      // [N,128]  (output 1)
    float* e_out = out + (size_t)N * HD;     // [E,128]  (output 2)

    // Workspace: Q, K, V node tables + softmax accumulators (5*N*128*4 B)
    float*    Qb    = (float*)d_ws;
    float*    Kb    = Qb + (size_t)N * HD;
    float*    Vb    = Kb + (size_t)N * HD;
    unsigned* mkey  = (unsigned*)(Vb + (size_t)N * HD);
    float*    denom = (float*)(mkey + (size_t)N * HD);

    hipMemsetAsync(wV,    0, (size_t)N * HD * sizeof(float),    stream);
    hipMemsetAsync(mkey,  0, (size_t)N * HD * sizeof(unsigned), stream); // key 0 < every real key
    hipMemsetAsync(denom, 0, (size_t)N * HD * sizeof(float),    stream);

    // --- Q/K/V projections (WMMA f32, A read once) ---
    dim3 blk(32 * WPB);
    const int nTilesN = (N + 15) / 16;       // N,E are multiples of 16
    const int nTilesE = (E + 15) / 16;
    dim3 gN((nTilesN + WPB - 1) / WPB);
    gemm128_bias_wmma<<<gN, blk, 0, stream>>>(h, Wq, bq, Qb, nTilesN);
    gemm128_bias_wmma<<<gN, blk, 0, stream>>>(h, Wk, bk, Kb, nTilesN);
    gemm128_bias_wmma<<<gN, blk, 0, stream>>>(h, Wv, bv, Vb, nTilesN);

    // --- fused e-projection + edge score + segment max ---
    dim3 gE((nTilesE + WPB - 1) / WPB);
    gemm_edge_score_fused<<<gE, blk, 0, stream>>>(e, We, be, Qb, Kb, src, dst,
                                                  e_out, mkey, nTilesE);

    // --- softmax denom + weighted scatter-sum ---
    dim3 eblk(256);                          // 8 waves / block
    dim3 egrid((E + 7) / 8);
    edge_denom_kernel<<<egrid, eblk, 0, stream>>>(e_out, mkey, dst, denom, E);
    edge_wv_kernel   <<<egrid, eblk, 0, stream>>>(e_out, mkey, denom, Vb, src, dst, wV, E);
}